// MyModel_87522843559452
// MI455X (gfx1250) — compile-verified
//
#include <hip/hip_runtime.h>

// ---------------------------------------------------------------------------
// Fused top-k retrieval for MI455X (gfx1250, wave32).
//   scores = Q[512,128] @ C[500000,128]^T  (f32, exact via V_WMMA_F32_16X16X4_F32)
//   top-100 per query, fused so the 1 GB score matrix never touches HBM.
//   Grid ordered so one 2 MB candidate chunk is shared through L2 by all 32
//   query-tile blocks -> candidates cross HBM ~once (256 MB, ~11 us floor).
// ---------------------------------------------------------------------------

typedef float v2f __attribute__((ext_vector_type(2)));
typedef float v8f __attribute__((ext_vector_type(8)));

constexpr int D_DIM = 128;   // embedding dim
constexpr int QTILE = 16;    // queries per block (== WMMA M == waves per block)
constexpr int SUB   = 1024;  // candidates scored per LDS panel
constexpr int SUBS_PER_CHUNK = 4;
constexpr int CHUNK = SUB * SUBS_PER_CHUNK;  // 4096 candidates per (block, chunk)
constexpr int TOPK  = 100;
constexpr float NEGF = -3.0e38f;

// ---------------------------------------------------------------------------
// Kernel 1: WMMA GEMM panel + in-register running top-100 per (query, chunk).
// Block: 512 threads = 16 waves. Wave w owns query row w of the 16-query tile.
// ---------------------------------------------------------------------------
__global__ void __launch_bounds__(512)
score_chunk_topk(const float* __restrict__ qmat,
                 const float* __restrict__ cmat,
                 int N, int nch,
                 float* __restrict__ wsV, int* __restrict__ wsI)
{
    __shared__ float s_scores[QTILE][SUB];   // 64 KB score panel (of 320 KB/WGP)

    const int tid   = threadIdx.x;
    const int lane  = tid & 31;
    const int wave  = tid >> 5;              // 0..15
    // blockIdx.x = query tile (fast dim) so the 32 blocks sharing one candidate
    // chunk are launched adjacently and hit the chunk in L2.
    const int qbase = blockIdx.x * QTILE;
    const int chunk = blockIdx.y;
    const long chunkBase = (long)chunk * CHUNK;

    // K-slice ownership per WMMA f32 fragment layout:
    //   VGPR0: lanes 0-15 -> K=k0+0, lanes 16-31 -> K=k0+2
    //   VGPR1: lanes 0-15 -> K=k0+1, lanes 16-31 -> K=k0+3
    const int koff = (lane & 16) ? 2 : 0;

    // ---- A fragments (query tile), cached in 64 VGPRs for the whole kernel.
    float aX[32], aY[32];
    {
        const float* qrow = qmat + (size_t)(qbase + (lane & 15)) * D_DIM + koff;
#pragma unroll
        for (int kk = 0; kk < 32; ++kk) {
            float2 t = *(const float2*)(qrow + kk * 4);
            aX[kk] = t.x; aY[kk] = t.y;
        }
    }

    // Running top-100 list for this wave's query: 4 slots/lane = 128 capacity.
    float cS0 = NEGF, cS1 = NEGF, cS2 = NEGF, cS3 = NEGF;
    int   cI0 = 0,    cI1 = 0,    cI2 = 0,    cI3 = 0;

    for (int s = 0; s < SUBS_PER_CHUNK; ++s) {
        const long subBase = chunkBase + (long)s * SUB;

        // ============== GEMM: 4 column tiles of 16 candidates per wave =====
#pragma unroll 1
        for (int i = 0; i < 4; ++i) {
            const int t = wave * 4 + i;            // 64 tiles / 16 waves
            long cr = subBase + t * 16 + (lane & 15);
            if (cr >= N) cr = N - 1;               // clamp; masked to -inf below
            const float* crow = cmat + (size_t)cr * D_DIM + koff;

            if (i < 3) {                           // prefetch next tile's row
                long crn = subBase + (t + 1) * 16 + (lane & 15);
                if (crn >= N) crn = N - 1;
                __builtin_prefetch(cmat + (size_t)crn * D_DIM + koff, 0, 1);
            }

            // Dual accumulators break the WMMA->WMMA RAW chain on C.
            v8f acc0 = {}, acc1 = {};
#pragma unroll
            for (int kk = 0; kk < 32; kk += 2) {   // D=128 in K=4 steps
                float2 b0 = *(const float2*)(crow + kk * 4);
                float2 b1 = *(const float2*)(crow + kk * 4 + 4);
                v2f a, b;
                a.x = aX[kk];     a.y = aY[kk];
                b.x = b0.x;       b.y = b0.y;
                acc0 = __builtin_amdgcn_wmma_f32_16x16x4_f32(
                         false, a, false, b, (short)0, acc0, false, false);
                a.x = aX[kk + 1]; a.y = aY[kk + 1];
                b.x = b1.x;       b.y = b1.y;
                acc1 = __builtin_amdgcn_wmma_f32_16x16x4_f32(
                         false, a, false, b, (short)0, acc1, false, false);
            }
            // C/D layout: VGPR r -> row r (lanes 0-15) / row r+8 (lanes 16-31)
            const int col = t * 16 + (lane & 15);
            const int r0  = (lane & 16) ? 8 : 0;
#pragma unroll
            for (int r = 0; r < 8; ++r) s_scores[r0 + r][col] = acc0[r] + acc1[r];
        }
        __syncthreads();

        // Mask out-of-range candidate columns (tail chunk only; uniform branch)
        if (subBase + SUB > N) {
            for (int idx = tid; idx < QTILE * SUB; idx += 512) {
                const int col = idx & (SUB - 1);
                if (subBase + col >= N) s_scores[idx >> 10][col] = NEGF;
            }
            __syncthreads();
        }

        // ====== merge-select: top-100 of (running list U 1024 new scores) ===
        // Threshold skip: once the list holds 100 entries, tau = entry 99
        // (slot 3, lane 3). If no new score beats tau, the top-100 is
        // unchanged -> skip all 100 argmax rounds (wave-uniform branch).
        bool doMerge = true;
        if (s > 0) {
            const float tau = __shfl(cS3, 3, 32);
            int cnt = 0;
#pragma unroll
            for (int j = 0; j < 32; ++j)
                cnt += (s_scores[wave][lane + 32 * j] > tau) ? 1 : 0;
#pragma unroll
            for (int off = 16; off > 0; off >>= 1)
                cnt += __shfl_xor(cnt, off, 32);
            doMerge = (cnt != 0);
        }

        if (doMerge) {
            float nS0 = NEGF, nS1 = NEGF, nS2 = NEGF, nS3 = NEGF;
            int   nI0 = 0,    nI1 = 0,    nI2 = 0,    nI3 = 0;
            for (int it = 0; it < TOPK; ++it) {
                // lane-local argmax over 32 LDS elements + 4 register slots
                float bs = NEGF; int bc = 0x7FFF;
#pragma unroll
                for (int j = 0; j < 32; ++j) {
                    const float v = s_scores[wave][lane + 32 * j];
                    if (v > bs) { bs = v; bc = lane + 32 * j; }
                }
                if (cS0 > bs) { bs = cS0; bc = 0x8000; }
                if (cS1 > bs) { bs = cS1; bc = 0x8001; }
                if (cS2 > bs) { bs = cS2; bc = 0x8002; }
                if (cS3 > bs) { bs = cS3; bc = 0x8003; }
                int code = (lane << 16) | bc;
                // wave-wide butterfly argmax (lowest code wins ties)
#pragma unroll
                for (int off = 16; off > 0; off >>= 1) {
                    const float os = __shfl_xor(bs, off, 32);
                    const int   oc = __shfl_xor(code, off, 32);
                    if (os > bs || (os == bs && oc < code)) { bs = os; code = oc; }
                }
                const int wl  = (code >> 16) & 31;
                const int low = code & 0xFFFF;
                int id;
                if (low & 0x8000) {                  // came from running list
                    const int slot = low & 3;
                    const int cid = (slot == 0) ? cI0 : (slot == 1) ? cI1
                                  : (slot == 2) ? cI2 : cI3;
                    id = __shfl(cid, wl, 32);
                    if (lane == wl) {
                        if      (slot == 0) cS0 = NEGF;
                        else if (slot == 1) cS1 = NEGF;
                        else if (slot == 2) cS2 = NEGF;
                        else                cS3 = NEGF;
                    }
                } else {                              // came from LDS panel
                    id = (int)(subBase + low);
                    if (lane == wl && low < SUB) s_scores[wave][low] = NEGF;
                }
                // k-th largest -> lane (it&31), slot (it>>5): sorted descending
                if (lane == (it & 31)) {
                    const int slot = it >> 5;
                    if      (slot == 0) { nS0 = bs; nI0 = id; }
                    else if (slot == 1) { nS1 = bs; nI1 = id; }
                    else if (slot == 2) { nS2 = bs; nI2 = id; }
                    else                { nS3 = bs; nI3 = id; }
                }
            }
            cS0 = nS0; cS1 = nS1; cS2 = nS2; cS3 = nS3;
            cI0 = nI0; cI1 = nI1; cI2 = nI2; cI3 = nI3;
        }
        __syncthreads();   // panel consumed; next GEMM overwrites it
    }

    // ---- emit exact per-(query, chunk) top-100 (descending) to workspace
    const size_t obase = ((size_t)(qbase + wave) * nch + chunk) * TOPK;
    int k;
    k = lane;      if (k < TOPK) { wsV[obase + k] = cS0; wsI[obase + k] = cI0; }
    k = 32 + lane; if (k < TOPK) { wsV[obase + k] = cS1; wsI[obase + k] = cI1; }
    k = 64 + lane; if (k < TOPK) { wsV[obase + k] = cS2; wsI[obase + k] = cI2; }
    k = 96 + lane; if (k < TOPK) { wsV[obase + k] = cS3; wsI[obase + k] = cI3; }
}

// ---------------------------------------------------------------------------
// Kernel 2: per query, exact top-100 of nch*100 (=12300) chunk partials.
// ---------------------------------------------------------------------------
constexpr int MAXM = 12320;   // >= ceil(500000/4096)*100 = 12300

__global__ void __launch_bounds__(256)
merge_topk(const float* __restrict__ wsV, const int* __restrict__ wsI,
           const int* __restrict__ ident,
           float* __restrict__ outV, int* __restrict__ outI, int M)
{
    __shared__ float s[MAXM];
    __shared__ float rS[8];
    __shared__ int   rI[8];

    const int q    = blockIdx.x;
    const int tid  = threadIdx.x;
    const int lane = tid & 31;
    const int wave = tid >> 5;

    const float* src = wsV + (size_t)q * M;
    for (int i = tid; i < M; i += 256) s[i] = src[i];
    __syncthreads();

    for (int it = 0; it < TOPK; ++it) {
        float bs = NEGF; int bi = 0x7FFFFFFF;
        for (int i = tid; i < M; i += 256) {
            const float v = s[i];
            if (v > bs) { bs = v; bi = i; }
        }
#pragma unroll
        for (int off = 16; off > 0; off >>= 1) {
            const float os = __shfl_xor(bs, off, 32);
            const int   oi = __shfl_xor(bi, off, 32);
            if (os > bs || (os == bs && oi < bi)) { bs = os; bi = oi; }
        }
        if (lane == 0) { rS[wave] = bs; rI[wave] = bi; }
        __syncthreads();
        if (wave == 0) {
            float vs = (lane < 8) ? rS[lane] : NEGF;
            int   vi = (lane < 8) ? rI[lane] : 0x7FFFFFFF;
#pragma unroll
            for (int off = 4; off > 0; off >>= 1) {
                const float os = __shfl_xor(vs, off, 32);
                const int   oi = __shfl_xor(vi, off, 32);
                if (os > vs || (os == vs && oi < vi)) { vs = os; vi = oi; }
            }
            if (lane == 0) { rS[0] = vs; rI[0] = vi; }
        }
        __syncthreads();
        if (tid == 0) {
            const int bestI = rI[0];
            outV[q * TOPK + it] = rS[0];
            outI[q * TOPK + it] = ident[wsI[(size_t)q * M + bestI]];
            s[bestI] = NEGF;
        }
        __syncthreads();
    }
}

// ---------------------------------------------------------------------------
extern "C" void kernel_launch(void* const* d_in, const int* in_sizes, int n_in,
                              void* d_out, int out_size, void* d_ws, size_t ws_size,
                              hipStream_t stream)
{
    const float* qmat  = (const float*)d_in[0];   // [Q, 128] f32
    const float* cmat  = (const float*)d_in[1];   // [N, 128] f32
    const int*   ident = (const int*)d_in[2];     // [N] int32
    (void)n_in; (void)out_size; (void)ws_size;

    const int Q   = in_sizes[0] / D_DIM;          // 512
    const int N   = in_sizes[1] / D_DIM;          // 500000
    const int nch = (N + CHUNK - 1) / CHUNK;      // 123
    const int M   = nch * TOPK;                   // 12300 (<= MAXM)

    // workspace: per-(query, chunk) top-100 scores then candidate indices
    float* wsV = (float*)d_ws;
    int*   wsI = (int*)(wsV + (size_t)Q * M);

    // outputs: values [Q,100] f32 followed by identifiers [Q,100] i32
    float* outV = (float*)d_out;
    int*   outI = (int*)(outV + (size_t)Q * TOPK);

    // qtile fast, chunk slow: blocks sharing a candidate chunk run adjacently
    dim3 g1(Q / QTILE, nch);
    hipLaunchKernelGGL(score_chunk_topk, g1, dim3(512), 0, stream,
                       qmat, cmat, N, nch, wsV, wsI);
    hipLaunchKernelGGL(merge_topk, dim3(Q), dim3(256), 0, stream,
                       wsV, wsI, ident, outV, outI, M);
}